// Corr2DM_44882408243584
// MI455X (gfx1250) — compile-verified
//
#include <hip/hip_runtime.h>

// CDNA5 (gfx1250) wave32 WMMA correlation layer.
// p[b,h,w,d] = sum_c x0[b,c,h,w]*x1[b,c,h,w+d-32]  (zero outside image)
// out[b,d,h,w] = (1/576) * 3x3 box sum of p over (h,w)
typedef __attribute__((ext_vector_type(2))) float v2f;
typedef __attribute__((ext_vector_type(8))) float v8f;

#define B_   4
#define C_   64
#define H_   256
#define W_   512
#define D_   65
#define HW_  (H_ * W_)
#define CHW_ (C_ * HW_)
#define DPAD 68   // padded d-stride in LDS to dodge bank conflicts
#define NWAVES 4

// One "group" = (row hh_i in {h-1,h,h+1}) x (A tile in {w0-16,w0,w0+16}).
// A fragments (16x64 f32, K=channels) are loaded once into 32 VGPRs, then the
// 5 banded B tiles (w' offsets -32..+32) are streamed through
// v_wmma_f32_16x16x4_f32 chains (16 per tile, K=64).
template <bool G>
__device__ __forceinline__ void corr_groups(const float* __restrict__ x0,
                                            const float* __restrict__ x1,
                                            float (*P)[48][DPAD],
                                            int b, int ho, int w0,
                                            int wave, int lane)
{
    const int m     = lane & 15;   // M row (A) / N col (B,C,D) this lane feeds
    const int khalf = lane >> 4;   // lanes 0-15: K={0,1}; lanes 16-31: K={2,3}

    for (int g = wave; g < 9; g += NWAVES) {
        const int hh_i = g / 3;
        const int wa_i = g % 3;
        const int hh   = ho - 1 + hh_i;
        const int wa   = w0 - 16 + 16 * wa_i;

        const bool rowOK = !G || ((hh >= 0) && (hh < H_));
        const int  wA    = wa + m;
        const bool aOK   = !G || (rowOK && (wA >= 0) && (wA < W_));

        // Base row pointer; clamped so addresses are always in-bounds even
        // when the value is masked to zero (border path only).
        const long rbase = (long)b * CHW_ + (long)(G ? (rowOK ? hh : 0) : hh) * W_;
        const float* pA  = x0 + rbase + (G ? (aOK ? wA : 0) : wA);

        // A fragments for the full K=64 chain: 16 x v2f = 32 VGPRs.
        v2f afrag[16];
#pragma unroll
        for (int kk = 0; kk < 16; ++kk) {
            const int k0 = kk * 4 + 2 * khalf;
            float a0 = pA[(long)k0 * HW_];
            float a1 = pA[(long)(k0 + 1) * HW_];
            if (G && !aOK) { a0 = 0.f; a1 = 0.f; }   // v_cndmask, no exec dance
            afrag[kk] = (v2f){a0, a1};
        }

        const int wl_base = 16 * wa_i;               // local w base in P

        for (int wb_i = 0; wb_i < 5; ++wb_i) {
            const int  wb  = wa + 16 * (wb_i - 2);
            const int  wB  = wb + m;
            const bool bOK = !G || (rowOK && (wB >= 0) && (wB < W_));
            const float* pB = x1 + rbase + (G ? (bOK ? wB : 0) : wB);

            v8f acc = {};
#pragma unroll
            for (int kk = 0; kk < 16; ++kk) {
                const int k0 = kk * 4 + 2 * khalf;
                float b0v = pB[(long)k0 * HW_];
                float b1v = pB[(long)(k0 + 1) * HW_];
                if (G && !bOK) { b0v = 0.f; b1v = 0.f; }
                v2f bv = {b0v, b1v};
                // 8 args: (neg_a, A, neg_b, B, c_mod, C, reuse_a, reuse_b)
                acc = __builtin_amdgcn_wmma_f32_16x16x4_f32(
                    false, afrag[kk], false, bv, (short)0, acc, false, false);
            }

            // Scatter tile into banded P. C/D layout: lanes 0-15 hold M=r,
            // lanes 16-31 hold M=r+8; N = lane&15. Each (w,d) slot is written
            // exactly once across the 45 tiles, so no init/atomics needed.
#pragma unroll
            for (int r = 0; r < 8; ++r) {
                const int mC = r + 8 * khalf;
                const int d  = (wb + m) - (wa + mC) + 32;
                if (d >= 0 && d < D_)
                    P[hh_i][wl_base + mC][d] = acc[r];
            }
        }
    }
}

__global__ __launch_bounds__(32 * NWAVES)
void corr2d_wmma_kernel(const float* __restrict__ x0,
                        const float* __restrict__ x1,
                        float* __restrict__ out)
{
    // P[row (h-1,h,h+1)][w local, covering w0-16..w0+31][displacement 0..64]
    __shared__ float P[3][48][DPAD];

    const int w0 = blockIdx.x * 16;   // 16 output columns per workgroup
    const int ho = blockIdx.y;        // one output row per workgroup
    const int b  = blockIdx.z;

    const int tid  = threadIdx.x;
    const int wave = tid >> 5;
    const int lane = tid & 31;

    // Interior blocks (~80%) need no bounds masking anywhere:
    // A cols in [w0-16, w0+31], B cols in [w0-48, w0+63], rows in [ho-1, ho+1].
    const bool interior = (w0 >= 48) && (w0 <= W_ - 64) &&
                          (ho >= 1) && (ho <= H_ - 2);
    if (interior) corr_groups<false>(x0, x1, P, b, ho, w0, wave, lane);
    else          corr_groups<true >(x0, x1, P, b, ho, w0, wave, lane);

    __syncthreads();

    // 3x3 box sum + normalization: 65*16 outputs per workgroup.
    for (int i = tid; i < D_ * 16; i += 32 * NWAVES) {
        const int d  = i >> 4;
        const int wl = i & 15;
        const int wc = wl + 16;                      // center in P's local w frame
        float s = 0.f;
#pragma unroll
        for (int r = 0; r < 3; ++r)
            s += P[r][wc - 1][d] + P[r][wc][d] + P[r][wc + 1][d];
        out[(((long)b * D_ + d) * H_ + ho) * W_ + (w0 + wl)] = s * (1.0f / 576.0f);
    }
}

extern "C" void kernel_launch(void* const* d_in, const int* in_sizes, int n_in,
                              void* d_out, int out_size, void* d_ws, size_t ws_size,
                              hipStream_t stream) {
    const float* x0 = (const float*)d_in[0];
    const float* x1 = (const float*)d_in[1];
    float* out = (float*)d_out;
    dim3 grid(W_ / 16, H_, B_);   // 32 x 256 x 4 workgroups, 128 threads (4 waves)
    corr2d_wmma_kernel<<<grid, dim3(32 * NWAVES), 0, stream>>>(x0, x1, out);
}